// DoubleTPKCBlock_35751307772131
// MI455X (gfx1250) — compile-verified
//
#include <hip/hip_runtime.h>
#include <hip/hip_bf16.h>
#include <stddef.h>

// ---------------------------------------------------------------------------
// Problem constants (from reference)
// ---------------------------------------------------------------------------
#define BB    2
#define FF    8
#define NIMG  16          // B*F
#define C1    16          // CIN
#define C2    32          // COUT
#define HH    128
#define WW    128
#define HW    16384       // H*W
#define NTAP  17          // center + 16 ring taps
#define LEAK  0.01f
#define EPSF  1e-5f

// Ring offsets in the exact order of _ring_offsets(RB=1,GB=1): dy=OX, dx=OY.
// tap 0 is the center (+sum-of-weights) tap.
__constant__ int c_ODY[NTAP] = {0, -2,-2,-2,-2,-2, -1, 0, 1,  2, 2, 2, 2, 2, -1, 0, 1};
__constant__ int c_ODX[NTAP] = {0, -2,-1, 0, 1, 2,  2, 2, 2, -2,-1, 0, 1, 2, -2,-2,-2};

// ---------------------------------------------------------------------------
// WMMA fp32 16x16x4 (exact fp32 math on the matrix pipe)
// ---------------------------------------------------------------------------
typedef float v2f __attribute__((ext_vector_type(2)));
typedef float v8f __attribute__((ext_vector_type(8)));

__device__ __forceinline__ v8f wmma_f32(v2f a, v2f b, v8f c) {
  // (neg_a, A, neg_b, B, c_mod, C, reuse_a, reuse_b)
  return __builtin_amdgcn_wmma_f32_16x16x4_f32(false, a, false, b, (short)0, c,
                                               false, false);
}

// ---------------------------------------------------------------------------
// CDNA5 async global->LDS copy (bypasses VGPRs, tracked on ASYNCcnt).
// GVS mode: mem_addr = SGPR64 base + VGPR i32 byte-offset. VDST = LDS byte
// address VGPR. Inline asm keeps it portable across ROCm7.2 / clang-23.
// ---------------------------------------------------------------------------
__device__ __forceinline__ void async_g2l_b32(const float* gbase,
                                              unsigned goff_bytes,
                                              float* lds_ptr) {
  unsigned lds = (unsigned)(size_t)lds_ptr;   // low 32 bits = LDS byte address
  asm volatile("global_load_async_to_lds_b32 %0, %1, %2"
               :
               : "v"(lds), "v"(goff_bytes), "s"(gbase)
               : "memory");
}

__device__ __forceinline__ void wait_asynccnt0() {
  asm volatile("s_wait_asynccnt 0x0" ::: "memory");
}

// ---------------------------------------------------------------------------
// Workspace layout (floats)
// ---------------------------------------------------------------------------
#define AOP1_FLTS (NTAP * 2 * 4 * 64)   //  8704 : [tap][mblk][kstep][r(2)][lane(32)]
#define AOP2_FLTS (NTAP * 2 * 8 * 64)   // 17408
#define AOP1_OFF  0
#define AOP2_OFF  AOP1_FLTS
#define ST1_OFF   (AOP1_FLTS + AOP2_FLTS)        // 64 floats: mu[32], rsig[32]
#define ST2_OFF   (ST1_OFF + 64)
#define Y1_OFF    32768
#define Y_FLTS    (NIMG * C2 * HW)               // 8388608 (32 MB)
#define Y2_OFF    (Y1_OFF + Y_FLTS)

// ---------------------------------------------------------------------------
// Kernel 1: rearrange weights into WMMA A-operand lane layout.
// A element (lane, r): M = lane&15 ; K = r + 2*(lane>>4)  (fp32 16x16x4 layout)
// tap 0 -> +sum_k W[o][c][k] ; tap t -> -W[o][c][t-1]
// ---------------------------------------------------------------------------
__global__ void prep_kernel(const float* __restrict__ W1,
                            const float* __restrict__ W2,
                            float* __restrict__ Aop1,
                            float* __restrict__ Aop2) {
  int tid = threadIdx.x;
  for (int idx = tid; idx < AOP1_FLTS; idx += 256) {
    int lane = idx & 31;
    int r    = (idx >> 5) & 1;
    int ks   = (idx >> 6) & 3;
    int mblk = (idx >> 8) & 1;
    int tt   = idx >> 9;
    int o = mblk * 16 + (lane & 15);
    int c = ks * 4 + r + 2 * (lane >> 4);
    float v;
    if (tt == 0) {
      v = 0.f;
      for (int t = 0; t < 16; ++t) v += W1[(o * C1 + c) * 16 + t];
    } else {
      v = -W1[(o * C1 + c) * 16 + (tt - 1)];
    }
    Aop1[idx] = v;
  }
  for (int idx = tid; idx < AOP2_FLTS; idx += 256) {
    int lane = idx & 31;
    int r    = (idx >> 5) & 1;
    int ks   = (idx >> 6) & 7;
    int mblk = (idx >> 9) & 1;
    int tt   = idx >> 10;
    int o = mblk * 16 + (lane & 15);
    int c = ks * 4 + r + 2 * (lane >> 4);
    float v;
    if (tt == 0) {
      v = 0.f;
      for (int t = 0; t < 16; ++t) v += W2[(o * C2 + c) * 16 + t];
    } else {
      v = -W2[(o * C2 + c) * 16 + (tt - 1)];
    }
    Aop2[idx] = v;
  }
}

// ---------------------------------------------------------------------------
// Kernel 2: layer-1 peak-conv via fp32 WMMA.
// Grid: (HH rows, NIMG images). Block: 128 (4 waves); wave w owns 32 pixels
// (two 16-px WMMA tiles) of row i, so each A operand feeds 2 WMMAs.
// Halo staged with async global->LDS; OOB slots zeroed via ds stores.
// x layout: (B, C1, F, H, W)
// ---------------------------------------------------------------------------
#define SX1_FLTS (C1 * 5 * 132)   // 10560

__global__ void conv1_kernel(const float* __restrict__ x,
                             const float* __restrict__ Aop,
                             const float* __restrict__ bias,
                             float* __restrict__ y1) {
  extern __shared__ float smem[];
  float* sA = smem;                 // AOP1_FLTS
  float* sX = smem + AOP1_FLTS;     // SX1_FLTS

  const int i = blockIdx.x;         // row
  const int n = blockIdx.y;         // image
  const int b = n >> 3, f = n & 7;
  const int tid = threadIdx.x;

  // Weights: plain async copy too (contiguous, always in range).
  for (int idx = tid; idx < AOP1_FLTS; idx += 128)
    async_g2l_b32(Aop, (unsigned)(idx * 4), &sA[idx]);

  // Image halo: async load in-range lanes, ds-store zeros for OOB lanes.
  for (int idx = tid; idx < SX1_FLTS; idx += 128) {
    int c   = idx / 660;
    int rem = idx - c * 660;
    int row = rem / 132;
    int col = rem - row * 132;
    int gi = i + row - 2;
    int gj = col - 2;
    if (gi >= 0 && gi < HH && gj >= 0 && gj < WW) {
      unsigned off = (unsigned)((((b * C1 + c) * FF + f) * HW + gi * WW + gj) * 4);
      async_g2l_b32(x, off, &sX[idx]);
    } else {
      sX[idx] = 0.f;
    }
  }
  wait_asynccnt0();
  __syncthreads();

  const int lane = tid & 31;
  const int wv   = tid >> 5;        // 0..3
  const int j0   = wv * 32;
  const int m    = lane & 15;
  const int half = lane >> 4;

  v8f acc0a = {0.f,0.f,0.f,0.f,0.f,0.f,0.f,0.f};
  v8f acc1a = {0.f,0.f,0.f,0.f,0.f,0.f,0.f,0.f};
  v8f acc0b = {0.f,0.f,0.f,0.f,0.f,0.f,0.f,0.f};
  v8f acc1b = {0.f,0.f,0.f,0.f,0.f,0.f,0.f,0.f};

  for (int tt = 0; tt < NTAP; ++tt) {
    const int row  = 2 + c_ODY[tt];
    const int colA = 2 + j0 + m + c_ODX[tt];
    const float* sAt = sA + tt * 512;          // 2 mblk * 4 ks * 64
#pragma unroll
    for (int ks = 0; ks < 4; ++ks) {
      const int c0 = ks * 4 + 2 * half;        // K = r + 2*half within group
      const float* sx0 = sX + (c0) * 660 + row * 132;
      const float* sx1 = sx0 + 660;
      v2f bopA; bopA.x = sx0[colA];      bopA.y = sx1[colA];
      v2f bopB; bopB.x = sx0[colA + 16]; bopB.y = sx1[colA + 16];
      const float* a0p = sAt + ks * 64;        // mblk 0
      const float* a1p = sAt + 256 + ks * 64;  // mblk 1
      v2f a0; a0.x = a0p[lane]; a0.y = a0p[32 + lane];
      v2f a1; a1.x = a1p[lane]; a1.y = a1p[32 + lane];
      acc0a = wmma_f32(a0, bopA, acc0a);
      acc0b = wmma_f32(a0, bopB, acc0b);
      acc1a = wmma_f32(a1, bopA, acc1a);
      acc1b = wmma_f32(a1, bopB, acc1b);
    }
  }

  float* ybA = y1 + (size_t)n * C2 * HW + (size_t)i * WW + (j0 + m);
#pragma unroll
  for (int r = 0; r < 8; ++r) {
    int o0 = r + 8 * half;
    int o1 = 16 + r + 8 * half;
    float bb0 = bias[o0], bb1 = bias[o1];
    ybA[(size_t)o0 * HW]      = acc0a[r] + bb0;
    ybA[(size_t)o0 * HW + 16] = acc0b[r] + bb0;
    ybA[(size_t)o1 * HW]      = acc1a[r] + bb1;
    ybA[(size_t)o1 * HW + 16] = acc1b[r] + bb1;
  }
}

// ---------------------------------------------------------------------------
// Kernel 3: per-channel BN statistics over (n, i, j) -> mu, rsqrt(var+eps)
// ---------------------------------------------------------------------------
__global__ void stats_kernel(const float* __restrict__ y, float* __restrict__ st) {
  const int ch = blockIdx.x;
  const int tid = threadIdx.x;
  float s = 0.f, s2 = 0.f;
  for (int nn = 0; nn < NIMG; ++nn) {
    const float* base = y + (size_t)(nn * C2 + ch) * HW;
    for (int p = tid; p < HW; p += 256) {
      float v = base[p];
      s += v; s2 += v * v;
    }
  }
  __shared__ float rs[256], rq[256];
  rs[tid] = s; rq[tid] = s2;
  __syncthreads();
  for (int off = 128; off > 0; off >>= 1) {
    if (tid < off) { rs[tid] += rs[tid + off]; rq[tid] += rq[tid + off]; }
    __syncthreads();
  }
  if (tid == 0) {
    const float N = (float)(NIMG * HW);
    float mu  = rs[0] / N;
    float var = rq[0] / N - mu * mu;
    st[ch]      = mu;
    st[32 + ch] = rsqrtf(var + EPSF);
  }
}

// ---------------------------------------------------------------------------
// Kernel 4: layer-2 peak-conv. BN1 + leaky applied while staging into LDS
// (zero padding applied AFTER BN, matching jnp.pad of the activated tensor),
// so staging keeps the VGPR path. Same 2-tiles-per-wave WMMA core, K=32.
// ---------------------------------------------------------------------------
#define SX2_FLTS (C2 * 5 * 132)   // 21120

__global__ void conv2_kernel(const float* __restrict__ y1,
                             const float* __restrict__ Aop,
                             const float* __restrict__ st1,
                             const float* __restrict__ g1,
                             const float* __restrict__ be1,
                             const float* __restrict__ bias,
                             float* __restrict__ y2) {
  extern __shared__ float smem[];
  float* sA = smem;                 // AOP2_FLTS
  float* sX = smem + AOP2_FLTS;     // SX2_FLTS

  const int i = blockIdx.x;
  const int n = blockIdx.y;
  const int tid = threadIdx.x;

  for (int idx = tid; idx < AOP2_FLTS; idx += 128)
    async_g2l_b32(Aop, (unsigned)(idx * 4), &sA[idx]);

  for (int idx = tid; idx < SX2_FLTS; idx += 128) {
    int c   = idx / 660;
    int rem = idx - c * 660;
    int row = rem / 132;
    int col = rem - row * 132;
    int gi = i + row - 2;
    int gj = col - 2;
    float v = 0.f;
    if (gi >= 0 && gi < HH && gj >= 0 && gj < WW) {
      float t = y1[(size_t)(n * C2 + c) * HW + gi * WW + gj];
      t = (t - st1[c]) * st1[32 + c] * g1[c] + be1[c];
      v = (t >= 0.f) ? t : LEAK * t;
    }
    sX[idx] = v;
  }
  wait_asynccnt0();
  __syncthreads();

  const int lane = tid & 31;
  const int wv   = tid >> 5;
  const int j0   = wv * 32;
  const int m    = lane & 15;
  const int half = lane >> 4;

  v8f acc0a = {0.f,0.f,0.f,0.f,0.f,0.f,0.f,0.f};
  v8f acc1a = {0.f,0.f,0.f,0.f,0.f,0.f,0.f,0.f};
  v8f acc0b = {0.f,0.f,0.f,0.f,0.f,0.f,0.f,0.f};
  v8f acc1b = {0.f,0.f,0.f,0.f,0.f,0.f,0.f,0.f};

  for (int tt = 0; tt < NTAP; ++tt) {
    const int row  = 2 + c_ODY[tt];
    const int colA = 2 + j0 + m + c_ODX[tt];
    const float* sAt = sA + tt * 1024;         // 2 mblk * 8 ks * 64
#pragma unroll
    for (int ks = 0; ks < 8; ++ks) {
      const int c0 = ks * 4 + 2 * half;
      const float* sx0 = sX + (c0) * 660 + row * 132;
      const float* sx1 = sx0 + 660;
      v2f bopA; bopA.x = sx0[colA];      bopA.y = sx1[colA];
      v2f bopB; bopB.x = sx0[colA + 16]; bopB.y = sx1[colA + 16];
      const float* a0p = sAt + ks * 64;        // mblk 0
      const float* a1p = sAt + 512 + ks * 64;  // mblk 1
      v2f a0; a0.x = a0p[lane]; a0.y = a0p[32 + lane];
      v2f a1; a1.x = a1p[lane]; a1.y = a1p[32 + lane];
      acc0a = wmma_f32(a0, bopA, acc0a);
      acc0b = wmma_f32(a0, bopB, acc0b);
      acc1a = wmma_f32(a1, bopA, acc1a);
      acc1b = wmma_f32(a1, bopB, acc1b);
    }
  }

  float* ybA = y2 + (size_t)n * C2 * HW + (size_t)i * WW + (j0 + m);
#pragma unroll
  for (int r = 0; r < 8; ++r) {
    int o0 = r + 8 * half;
    int o1 = 16 + r + 8 * half;
    float bb0 = bias[o0], bb1 = bias[o1];
    ybA[(size_t)o0 * HW]      = acc0a[r] + bb0;
    ybA[(size_t)o0 * HW + 16] = acc0b[r] + bb0;
    ybA[(size_t)o1 * HW]      = acc1a[r] + bb1;
    ybA[(size_t)o1 * HW + 16] = acc1b[r] + bb1;
  }
}

// ---------------------------------------------------------------------------
// Kernel 5: BN2 + leaky + mean over the 8 frames -> (B, C2, 1, H, W)
// ---------------------------------------------------------------------------
__global__ void finalize_kernel(const float* __restrict__ y2,
                                const float* __restrict__ st2,
                                const float* __restrict__ g2,
                                const float* __restrict__ be2,
                                float* __restrict__ out) {
  int idx = blockIdx.x * 256 + threadIdx.x;   // 2*32*16384 total
  int b   = idx >> 19;                        // / (32*16384)
  int rem = idx & ((1 << 19) - 1);
  int o   = rem >> 14;
  int p   = rem & (HW - 1);
  float mu = st2[o], rsig = st2[32 + o], g = g2[o], be = be2[o];
  float s = 0.f;
#pragma unroll
  for (int f = 0; f < FF; ++f) {
    float v = y2[(size_t)((b * FF + f) * C2 + o) * HW + p];
    v = (v - mu) * rsig * g + be;
    v = (v >= 0.f) ? v : LEAK * v;
    s += v;
  }
  out[idx] = s * 0.125f;
}

// ---------------------------------------------------------------------------
// Host launcher
// ---------------------------------------------------------------------------
extern "C" void kernel_launch(void* const* d_in, const int* in_sizes, int n_in,
                              void* d_out, int out_size, void* d_ws, size_t ws_size,
                              hipStream_t stream) {
  (void)in_sizes; (void)n_in; (void)out_size; (void)ws_size;
  const float* x   = (const float*)d_in[0];
  const float* W1  = (const float*)d_in[1];
  const float* b1  = (const float*)d_in[2];
  const float* g1  = (const float*)d_in[3];
  const float* be1 = (const float*)d_in[4];
  const float* W2  = (const float*)d_in[5];
  const float* b2  = (const float*)d_in[6];
  const float* g2  = (const float*)d_in[7];
  const float* be2 = (const float*)d_in[8];

  float* ws   = (float*)d_ws;                 // needs ~68 MB
  float* Aop1 = ws + AOP1_OFF;
  float* Aop2 = ws + AOP2_OFF;
  float* st1  = ws + ST1_OFF;
  float* st2  = ws + ST2_OFF;
  float* y1   = ws + Y1_OFF;
  float* y2   = ws + Y2_OFF;

  prep_kernel<<<1, 256, 0, stream>>>(W1, W2, Aop1, Aop2);

  dim3 cgrid(HH, NIMG);
  size_t sh1 = (size_t)(AOP1_FLTS + SX1_FLTS) * sizeof(float); //  77 KB
  conv1_kernel<<<cgrid, 128, sh1, stream>>>(x, Aop1, b1, y1);

  stats_kernel<<<C2, 256, 0, stream>>>(y1, st1);

  size_t sh2 = (size_t)(AOP2_FLTS + SX2_FLTS) * sizeof(float); // 154 KB (<=320 KB WGP LDS)
  conv2_kernel<<<cgrid, 128, sh2, stream>>>(y1, Aop2, st1, g1, be1, b2, y2);

  stats_kernel<<<C2, 256, 0, stream>>>(y2, st2);

  finalize_kernel<<<(BB * C2 * HW) / 256, 256, 0, stream>>>(y2, st2, g2, be2,
                                                            (float*)d_out);
}